// IPSMeanAggregator_9320079032904
// MI455X (gfx1250) — compile-verified
//
#include <hip/hip_runtime.h>
#include <math.h>

// ---------------------------------------------------------------------------
// IPS top-k mean aggregator, gather formulation.
//   traffic: rel_pos 16MB + mask 4MB + <=3 gathered z rows (12MB) + out 4MB
//            ~= 36MB  vs 256MB for the dense einsum  ->  ~1.6us HBM floor.
// wave32: one wave per batch row, 2 neighbors per lane.
// CDNA5 paths: global_load_async_to_lds_b128 (ASYNCcnt) staging of rel_pos
// via inline asm (portable across toolchains, per CDNA5_HIP.md), and
// ds_bpermute-based __shfl_xor butterflies for the wave-wide top-3 argmax.
// ---------------------------------------------------------------------------

#define ROWS_PER_BLOCK 8
#define NNB 64
#define DIM 64

#if defined(__HIP_DEVICE_COMPILE__)
#define IPS_ASYNC_LDS 1
typedef __attribute__((address_space(3))) void l_void;
#else
#define IPS_ASYNC_LDS 0
#endif

__device__ __forceinline__ float ips_score(float4 rp) {
  // matches reference _interaction_priority_scores for a valid (mask==1) lane
  const float dx = rp.x, dy = rp.y, dvx = rp.z, dvy = rp.w;
  const float d2   = fmaxf(dx * dx + dy * dy, 1e-12f);
  const float dist = sqrtf(d2);
  const float closing = fmaxf(-(dx * dvx + dy * dvy) / dist, 0.0f);
  const float ttc = (closing > 1e-6f) ? (dist / (closing + 1e-6f)) : 1e6f;
  return 1.0f / (dist + 1e-3f) + 0.5f * closing + 1.0f / (ttc + 1e-3f);
}

__global__ __launch_bounds__(ROWS_PER_BLOCK * 32) void
ips_topk_mean_kernel(const float* __restrict__ zs,    // [B, N, D]
                     const float* __restrict__ rel,   // [B, N, 4]
                     const int* __restrict__ mask,    // [B, N]
                     float* __restrict__ out,         // [B, D]
                     int Brows) {
  const int wave = threadIdx.x >> 5;
  const int lane = threadIdx.x & 31;
  const int row  = blockIdx.x * ROWS_PER_BLOCK + wave;
  if (row >= Brows) return;

  const float NEG_INF = -__builtin_inff();

  // ---- this row's rel_pos (64 x float4); lane owns neighbors 2L and 2L+1 --
  const float4* grel = (const float4*)rel + (size_t)row * NNB;
  float4 rp0, rp1;

#if IPS_ASYNC_LDS
  __shared__ float4 srel[ROWS_PER_BLOCK * NNB];  // 8 KB staged tile
  const int sbase = wave * NNB + lane * 2;
  // LDS byte offsets (AS3 pointers are 32-bit LDS offsets on amdgcn)
  const unsigned lds0 =
      (unsigned)(unsigned long)(l_void*)(&srel[sbase]);
  const unsigned lds1 = lds0 + 16u;
  const unsigned long long ga0 = (unsigned long long)(const void*)(grel + lane * 2);
  const unsigned long long ga1 = ga0 + 16ull;
  // async DMA global -> LDS, tracked on per-wave ASYNCcnt; wait, then consume.
  asm volatile(
      "global_load_async_to_lds_b128 %0, %1, off\n\t"
      "global_load_async_to_lds_b128 %2, %3, off\n\t"
      "s_wait_asynccnt 0"
      :
      : "v"(lds0), "v"(ga0), "v"(lds1), "v"(ga1)
      : "memory");
  rp0 = srel[sbase];
  rp1 = srel[sbase + 1];
#else
  rp0 = grel[lane * 2];
  rp1 = grel[lane * 2 + 1];
#endif

  const int2 m2 = *(const int2*)(mask + (size_t)row * NNB + lane * 2);

  // masked scores (valid scores finite >= 0, invalid -> -inf)
  float s0 = (m2.x > 0) ? ips_score(rp0) * (float)m2.x : NEG_INF;
  float s1 = (m2.y > 0) ? ips_score(rp1) * (float)m2.y : NEG_INF;
  const int n0 = lane * 2;
  const int n1 = lane * 2 + 1;

  // ---- top-3 via 3 wave-wide argmax butterflies (tie -> lower index) ------
  int win0 = 0, win1 = 0, win2 = 0;
  int keff = 0;
#pragma unroll
  for (int p = 0; p < 3; ++p) {
    // per-lane best of its 2 slots (slot0 has the lower index)
    float bs = (s1 > s0) ? s1 : s0;
    int   bn = (s1 > s0) ? n1 : n0;
#pragma unroll
    for (int off = 16; off > 0; off >>= 1) {
      const float os = __shfl_xor(bs, off, 32);
      const int   on = __shfl_xor(bn, off, 32);
      if (os > bs || (os == bs && on < bn)) { bs = os; bn = on; }
    }
    if (bs == NEG_INF) break;  // fewer than p+1 valid neighbors
    if (p == 0) win0 = bn; else if (p == 1) win1 = bn; else win2 = bn;
    keff = p + 1;
    // knock the winner out (exactly one lane owns index bn)
    if (bn == n0) s0 = NEG_INF;
    if (bn == n1) s1 = NEG_INF;
  }

  // ---- mean of selected z-vectors: lane owns components d=2*lane, 2*lane+1
  const float* zrow = zs + (size_t)row * (NNB * DIM);
  const int d = lane * 2;
  float ax = 0.0f, ay = 0.0f;
  if (keff > 0) { const float2 v = *(const float2*)(zrow + win0 * DIM + d); ax += v.x; ay += v.y; }
  if (keff > 1) { const float2 v = *(const float2*)(zrow + win1 * DIM + d); ax += v.x; ay += v.y; }
  if (keff > 2) { const float2 v = *(const float2*)(zrow + win2 * DIM + d); ax += v.x; ay += v.y; }

  const float inv = (keff > 0) ? (1.0f / (float)keff) : 0.0f;
  float2 o;
  o.x = ax * inv;
  o.y = ay * inv;
  *(float2*)(out + (size_t)row * DIM + d) = o;
}

extern "C" void kernel_launch(void* const* d_in, const int* in_sizes, int n_in,
                              void* d_out, int out_size, void* d_ws, size_t ws_size,
                              hipStream_t stream) {
  // inputs (setup_inputs order): 0=ego_z [B,64] (unused), 1=neighbor_zs [B,N,D],
  //                              2=neighbor_rel_pos [B,N,4], 3=mask [B,N]
  const float* zs   = (const float*)d_in[1];
  const float* rel  = (const float*)d_in[2];
  const int*   mask = (const int*)d_in[3];
  float* out = (float*)d_out;

  const int B = in_sizes[0] / DIM;  // ego_z is [B, INTENT_DIM]
  const int grid = (B + ROWS_PER_BLOCK - 1) / ROWS_PER_BLOCK;
  ips_topk_mean_kernel<<<grid, ROWS_PER_BLOCK * 32, 0, stream>>>(zs, rel, mask, out, B);
}